// fwd_LSTM_146028888507
// MI455X (gfx1250) — compile-verified
//
#include <hip/hip_runtime.h>
#include <math.h>

// CDNA5 / gfx1250: wave32, fp32 WMMA (V_WMMA_F32_16X16X4_F32) + async global->LDS DMA.
typedef __attribute__((ext_vector_type(2))) float v2f;
typedef __attribute__((ext_vector_type(8))) float v8f;
typedef __attribute__((ext_vector_type(4))) int   v4i;

#define LDSPAD 68  // padded LDS row stride in floats (272B: 16B-aligned, avoids 64-bank conflicts)

#ifndef __has_builtin
#define __has_builtin(x) 0
#endif

#if __has_builtin(__builtin_amdgcn_global_load_async_to_lds_b128)
#define HAVE_ASYNC_LDS 1
#else
#define HAVE_ASYNC_LDS 0
#endif

#define AS1 __attribute__((address_space(1)))
#define AS3 __attribute__((address_space(3)))

__device__ __forceinline__ void wait_async_all() {
#if HAVE_ASYNC_LDS
#if __has_builtin(__builtin_amdgcn_s_wait_asynccnt)
    __builtin_amdgcn_s_wait_asynccnt(0);
#else
    asm volatile("s_wait_asynccnt 0x0" ::: "memory");
#endif
#endif
}

__device__ __forceinline__ void async_b128(const float* g, float* l) {
#if HAVE_ASYNC_LDS
    // Signature (from clang diagnostic): (v4i addrspace(1)*, v4i addrspace(3)*, imm offset, imm cpol)
    __builtin_amdgcn_global_load_async_to_lds_b128((AS1 v4i*)g, (AS3 v4i*)l, 0, 0);
#else
    *(float4*)l = *(const float4*)g;
#endif
}

__device__ __forceinline__ float sigmoidf_(float x) {
    return 1.0f / (1.0f + __expf(-x));
}

// Fused LSTM cell: gates = x @ Wih^T + h_in @ Whh^T + bih + bhh, then
// c = sig(f)*c_in + sig(i)*tanh(g); h = sig(o)*tanh(c).
// Grid: (B/16, S * Hl/16). Block: 128 threads = 4 wave32s, wave w computes gate w.
// Double-buffered LDS tiles staged with async global->LDS DMA, overlapped with WMMA.
extern "C" __global__ void __launch_bounds__(128)
lstm_cell_wmma(const float* __restrict__ X, int ldx,
               const float* __restrict__ Hin,   // [B, S/2, Hl] or nullptr (level 1)
               const float* __restrict__ Cin,   // [B, S/2, Hl] or nullptr (level 1)
               const float* __restrict__ Wih,   // [4*Hl, Dl] row-major
               const float* __restrict__ Whh,   // [4*Hl, Hl] row-major
               const float* __restrict__ bih,
               const float* __restrict__ bhh,
               float* __restrict__ Hout,        // [B, S, Hl] or nullptr
               float* __restrict__ Cout,        // [B, S, Hl]
               int S, int Dl, int Hl, int nTiles)
{
    __shared__ float lds_x[2][16 * LDSPAD];       // A tile: 16 batch rows x 64 k (double buffered)
    __shared__ float lds_w[2][4 * 16 * LDSPAD];   // B tile per gate: 16 cols x 64 k

    const int tid  = threadIdx.x;
    const int lane = tid & 31;
    const int wv   = tid >> 5;                    // gate index 0..3 (i,f,g,o)
    const int rb16 = blockIdx.x * 16;             // batch-row block
    const int gy   = blockIdx.y;
    const int s    = gy / nTiles;                 // sequence step
    const int col0 = (gy % nTiles) * 16;          // hidden-column block

    const int nx   = Dl >> 6;                     // 64-wide k-tiles in the x phase
    const int ntot = nx + (Hin ? (Hl >> 6) : 0);  // level 1: h==0 -> skip Whh GEMM entirely

    // Stage one 16x64 A tile (cooperative) + four 16x64 B tiles (one per wave) into buffer pb.
    auto stage = [&](int t, int pb) {
        const bool isx  = t < nx;
        const int  k0   = (isx ? t : t - nx) << 6;
        const float* src = isx ? X : Hin;
        const long rstr = isx ? (long)ldx : (long)(S >> 1) * Hl;
        const long coff = isx ? (long)s * Dl : (long)(s >> 1) * Hl;
        const float* W  = isx ? Wih : Whh;
        const long wld  = isx ? (long)Dl : (long)Hl;

        // A tile: 256 x b128, 2 per thread
        {
            int e = tid;
#pragma unroll
            for (int i = 0; i < 2; ++i, e += 128) {
                const int r = e >> 4, c4 = (e & 15) << 2;
                async_b128(src + (long)(rb16 + r) * rstr + coff + k0 + c4,
                           &lds_x[pb][r * LDSPAD + c4]);
            }
        }
        // This wave's gate B tile: 256 x b128, 8 per lane
        {
            int e = lane;
#pragma unroll
            for (int i = 0; i < 8; ++i, e += 32) {
                const int r = e >> 4, c4 = (e & 15) << 2;
                async_b128(W + (long)(wv * Hl + col0 + r) * wld + k0 + c4,
                           &lds_w[pb][(wv * 16 + r) * LDSPAD + c4]);
            }
        }
    };

    v8f acc = {};                                 // 16x16 f32 accumulator (8 VGPRs)

    // Fragment addressing per ISA 7.12.2:
    //  A (16x4): lane -> M = lane%16, K = (lane/16)*2 + vgpr
    //  B (4x16): lane -> N = lane%16, K = (lane/16)*2 + vgpr
    const int mn   = lane & 15;
    const int koff = (lane >> 4) << 1;

    // Software pipeline: DMA of tile t+1 overlaps WMMA of tile t.
    stage(0, 0);
    wait_async_all();
    __syncthreads();

    for (int t = 0; t < ntot; ++t) {
        if (t + 1 < ntot) stage(t + 1, (t + 1) & 1);

        const float* lx = &lds_x[t & 1][mn * LDSPAD + koff];
        const float* lw = &lds_w[t & 1][(wv * 16 + mn) * LDSPAD + koff];
#pragma unroll
        for (int kk = 0; kk < 64; kk += 4) {
            v2f a, b;
            a.x = lx[kk]; a.y = lx[kk + 1];
            b.x = lw[kk]; b.y = lw[kk + 1];
            acc = __builtin_amdgcn_wmma_f32_16x16x4_f32(
                /*neg_a=*/false, a, /*neg_b=*/false, b,
                /*c_mod=*/(short)0, acc, /*reuse_a=*/false, /*reuse_b=*/false);
        }
        wait_async_all();
        __syncthreads();
    }

    // Bias add: depends only on (gate, column n)
    {
        const int n = lane & 15;
        const float bias = bih[wv * Hl + col0 + n] + bhh[wv * Hl + col0 + n];
#pragma unroll
        for (int r = 0; r < 8; ++r) acc[r] += bias;
    }

    // Exchange the four gate tiles through LDS (reuse lds_w[0]; final barrier already passed)
    float* lg = &lds_w[0][0];
#pragma unroll
    for (int r = 0; r < 8; ++r) lg[wv * 256 + r * 32 + lane] = acc[r];
    __syncthreads();

    // Combine i,f,g,o -> c,h for the 16x16 tile (256 elems / 128 threads)
    for (int e = tid; e < 256; e += 128) {
        const int r = e >> 5, l = e & 31;
        const int m = r + ((l >> 4) << 3);        // C-layout: VGPR r, lanes16-31 -> M=8+r
        const int n = l & 15;
        const float iv = lg[0 * 256 + e];
        const float fv = lg[1 * 256 + e];
        const float gv = lg[2 * 256 + e];
        const float ov = lg[3 * 256 + e];
        const long  b  = rb16 + m;
        float cin = 0.0f;
        if (Cin) cin = Cin[(b * (long)(S >> 1) + (s >> 1)) * Hl + col0 + n];
        const float cnew = sigmoidf_(fv) * cin + sigmoidf_(iv) * tanhf(gv);
        const long oidx = (b * (long)S + s) * Hl + col0 + n;
        Cout[oidx] = cnew;
        if (Hout) Hout[oidx] = sigmoidf_(ov) * tanhf(cnew);
    }
}

// conv1d(1->1, k=3, stride=2, pad=1, no bias) + softmax over the output row.
// One block per (batch,row-step); L in {2048,1024,512} -> nout in {1024,512,256}.
extern "C" __global__ void __launch_bounds__(256)
conv_softmax(const float* __restrict__ in, const float* __restrict__ w3,
             float* __restrict__ out, int L)
{
    __shared__ float sv[1024];
    __shared__ float sr[256];
    const int row  = blockIdx.x;
    const int tid  = threadIdx.x;
    const int nout = L >> 1;
    const float w0 = w3[0], w1 = w3[1], w2 = w3[2];
    const float* x = in + (long)row * L;

    float lmax = -3.4e38f;
    for (int j = tid; j < nout; j += 256) {
        const float xm1 = (j == 0) ? 0.0f : x[2 * j - 1];
        const float v = w0 * xm1 + w1 * x[2 * j] + w2 * x[2 * j + 1];
        sv[j] = v;
        lmax = fmaxf(lmax, v);
    }
    sr[tid] = lmax;
    __syncthreads();
    for (int off = 128; off > 0; off >>= 1) {
        if (tid < off) sr[tid] = fmaxf(sr[tid], sr[tid + off]);
        __syncthreads();
    }
    const float mx = sr[0];
    __syncthreads();

    float lsum = 0.0f;
    for (int j = tid; j < nout; j += 256) {
        const float e = __expf(sv[j] - mx);
        sv[j] = e;
        lsum += e;
    }
    sr[tid] = lsum;
    __syncthreads();
    for (int off = 128; off > 0; off >>= 1) {
        if (tid < off) sr[tid] += sr[tid + off];
        __syncthreads();
    }
    const float inv = 1.0f / sr[0];
    for (int j = tid; j < nout; j += 256) {
        out[(long)row * nout + j] = sv[j] * inv;
    }
}

extern "C" void kernel_launch(void* const* d_in, const int* in_sizes, int n_in,
                              void* d_out, int out_size, void* d_ws, size_t ws_size,
                              hipStream_t stream)
{
    (void)in_sizes; (void)n_in; (void)out_size; (void)ws_size;
    const int B = 4096, D = 2048, H = 2048;

    const float* spec = (const float*)d_in[0];
    const float* Wih[4] = {(const float*)d_in[1],  (const float*)d_in[5],
                           (const float*)d_in[9],  (const float*)d_in[13]};
    const float* Whh[4] = {(const float*)d_in[2],  (const float*)d_in[6],
                           (const float*)d_in[10], (const float*)d_in[14]};
    const float* bih[4] = {(const float*)d_in[3],  (const float*)d_in[7],
                           (const float*)d_in[11], (const float*)d_in[15]};
    const float* bhh[4] = {(const float*)d_in[4],  (const float*)d_in[8],
                           (const float*)d_in[12], (const float*)d_in[16]};
    const float* w12h = (const float*)d_in[17];
    const float* w12c = (const float*)d_in[18];
    const float* w23h = (const float*)d_in[19];
    const float* w23c = (const float*)d_in[20];
    const float* w34h = (const float*)d_in[21];

    // Workspace layout (floats): h | c | th | tc  (~101 MB total, reused per level)
    float* hbuf  = (float*)d_ws;
    float* cbuf  = hbuf  + (size_t)B * H;
    float* thbuf = cbuf  + (size_t)B * H;
    float* tcbuf = thbuf + (size_t)B * (H / 2);

    const dim3 blk(128);
    const dim3 tblk(256);

    // ---- Level 1: S=1, Dl=2048, Hl=2048; zero init state -> skip Whh GEMM ----
    {
        const int S = 1, Dl = 2048, Hl = 2048;
        dim3 grid(B / 16, S * Hl / 16);
        lstm_cell_wmma<<<grid, blk, 0, stream>>>(spec, D, nullptr, nullptr,
            Wih[0], Whh[0], bih[0], bhh[0], hbuf, cbuf, S, Dl, Hl, Hl / 16);
        conv_softmax<<<dim3(B * S), tblk, 0, stream>>>(hbuf, w12h, thbuf, Hl);
        conv_softmax<<<dim3(B * S), tblk, 0, stream>>>(cbuf, w12c, tcbuf, Hl);
    }
    // ---- Level 2: S=2, Dl=1024, Hl=1024 ----
    {
        const int S = 2, Dl = 1024, Hl = 1024;
        dim3 grid(B / 16, S * Hl / 16);
        lstm_cell_wmma<<<grid, blk, 0, stream>>>(spec, D, thbuf, tcbuf,
            Wih[1], Whh[1], bih[1], bhh[1], hbuf, cbuf, S, Dl, Hl, Hl / 16);
        conv_softmax<<<dim3(B * S), tblk, 0, stream>>>(hbuf, w23h, thbuf, Hl);
        conv_softmax<<<dim3(B * S), tblk, 0, stream>>>(cbuf, w23c, tcbuf, Hl);
    }
    // ---- Level 3: S=4, Dl=512, Hl=512 (reference uses w34h for BOTH trans) ----
    {
        const int S = 4, Dl = 512, Hl = 512;
        dim3 grid(B / 16, S * Hl / 16);
        lstm_cell_wmma<<<grid, blk, 0, stream>>>(spec, D, thbuf, tcbuf,
            Wih[2], Whh[2], bih[2], bhh[2], hbuf, cbuf, S, Dl, Hl, Hl / 16);
        conv_softmax<<<dim3(B * S), tblk, 0, stream>>>(hbuf, w34h, thbuf, Hl);
        conv_softmax<<<dim3(B * S), tblk, 0, stream>>>(cbuf, w34h, tcbuf, Hl);
    }
    // ---- Level 4: S=8, Dl=256, Hl=256; output = c4 reshape -> write d_out directly ----
    {
        const int S = 8, Dl = 256, Hl = 256;
        dim3 grid(B / 16, S * Hl / 16);
        lstm_cell_wmma<<<grid, blk, 0, stream>>>(spec, D, thbuf, tcbuf,
            Wih[3], Whh[3], bih[3], bhh[3], nullptr, (float*)d_out, S, Dl, Hl, Hl / 16);
    }
}